// RPNLoss_77129022701996
// MI455X (gfx1250) — compile-verified
//
#include <hip/hip_runtime.h>
#include <cstdint>

// ---------------- problem constants ----------------
#define B_ 4
#define A_ 3
#define H_ 168
#define W_ 256
#define M_ 32
#define HW_ (H_*W_)
#define N_ (HW_*A_)          // 129024 anchors per image
#define NUM_PER_IMAGE 256
#define NUM_POS_MAX 128
#define FG_THR 0.7f
#define BG_THR 0.3f
#define BETA (1.0f/9.0f)

typedef __attribute__((ext_vector_type(2))) float        v2f;
typedef __attribute__((ext_vector_type(8))) float        v8f;
typedef __attribute__((ext_vector_type(4))) unsigned int u32x4;
typedef __attribute__((ext_vector_type(8))) int          i32x8;
typedef __attribute__((ext_vector_type(4))) int          i32x4;

// ---------------- TDM: DMA the 32x4 f32 target tile into LDS ----------------
// 2D descriptor: 128 contiguous 4-byte elements (one image's targets).
__device__ __forceinline__ void tdm_load_targets(const float* gsrc, void* lds_dst) {
  unsigned lds_addr = (unsigned)(uintptr_t)lds_dst;           // low 32 bits = LDS offset
  uint64_t ga = (uint64_t)(uintptr_t)gsrc;
  u32x4 g0;
  g0[0] = 1u;                                                 // count=1, user mode, no gather
  g0[1] = lds_addr;                                           // lds_addr[31:0]
  g0[2] = (unsigned)(ga & 0xFFFFFFFFu);                       // global_addr[31:0]
  g0[3] = (unsigned)((ga >> 32) & 0x1FFFFFFu) | (2u << 30);   // global_addr[56:32], type=2
  i32x8 g1;
  g1[0] = (2 << 16);          // workgroup_mask=0 (not in cluster), data_size=2 (4B)
  g1[1] = (128 << 16);        // tensor_dim0[15:0]=128
  g1[2] = (1 << 16);          // tensor_dim0[31:16]=0 | tensor_dim1[15:0]=1
  g1[3] = (128 << 16);        // tensor_dim1[31:16]=0 | tile_dim0=128
  g1[4] = 1;                  // tile_dim1=1 | tile_dim2=0
  g1[5] = 128;                // tensor_dim0_stride[31:0]=128
  g1[6] = (128 << 16);        // dim0_stride[47:32]=0 | dim1_stride[15:0]=128
  g1[7] = 0;                  // dim1_stride[47:16]=0
  i32x4 gz4  = {0, 0, 0, 0};                  // groups 2/3 unused (<=2D tensor)
  i32x8 gz8  = {0, 0, 0, 0, 0, 0, 0, 0};      // extra group (6-arg toolchain)
  __builtin_amdgcn_tensor_load_to_lds(g0, g1, gz4, gz4, gz8, 0);
  __builtin_amdgcn_s_wait_tensorcnt(0);
}

__device__ __forceinline__ float iou_one(float4 a, float4 t) {
  float xtl = fmaxf(a.x, t.x), ytl = fmaxf(a.y, t.y);
  float xrb = fminf(a.z, t.z), yrb = fminf(a.w, t.w);
  float inter = fmaxf(xrb - xtl + 1.0f, 0.0f) * fmaxf(yrb - ytl + 1.0f, 0.0f);
  float a1 = (a.z - a.x + 1.0f) * (a.w - a.y + 1.0f);
  float a2 = (t.z - t.x + 1.0f) * (t.w - t.y + 1.0f);
  return inter / (a1 + a2 - inter);
}

// ---------------- kernel 0: init workspace ----------------
__global__ void rpn_init_kernel(int* colmax, int* pos_cnt, int* neg_cnt, float* accum) {
  int t = threadIdx.x;
  if (t < B_ * M_) colmax[t] = 0;                 // IoU >= 0 -> 0.0f bits == 0
  if (t < B_) { pos_cnt[t] = 0; neg_cnt[t] = 0; }
  if (t < 3) accum[t] = 0.0f;
}

// ---------------- kernel 1: per-target column max of IoU ----------------
__global__ void rpn_colmax_kernel(const float* __restrict__ anchors,
                                  const float* __restrict__ targets,
                                  int* __restrict__ colmax) {
  int b = blockIdx.y;
  int n = blockIdx.x * blockDim.x + threadIdx.x;
  __shared__ float4 tgt[M_];
  __shared__ int    cmax_s[M_];
  if (threadIdx.x < M_) cmax_s[threadIdx.x] = 0;
  if (threadIdx.x < 32) tdm_load_targets(targets + (size_t)b * M_ * 4, tgt);
  __syncthreads();

  float4 a = ((const float4*)anchors)[(size_t)b * N_ + n];
  for (int m = 0; m < M_; ++m) {
    float iou = iou_one(a, tgt[m]);
    atomicMax(&cmax_s[m], __float_as_int(iou));   // ds_max_i32; valid for non-neg floats
  }
  __syncthreads();
  if (threadIdx.x < M_)
    atomicMax(&colmax[b * M_ + threadIdx.x], cmax_s[threadIdx.x]);
}

// ---------------- kernel 2: match + filter + sample ----------------
__global__ void rpn_assign_kernel(const float* __restrict__ anchors,
                                  const float* __restrict__ targets,
                                  const float* __restrict__ sizes,
                                  const int* __restrict__ colmax,
                                  int* __restrict__ pos_cnt, int* __restrict__ neg_cnt,
                                  int* __restrict__ pos_buf, int* __restrict__ neg_buf) {
  int b = blockIdx.y;
  int n = blockIdx.x * blockDim.x + threadIdx.x;
  __shared__ float4 tgt[M_];
  __shared__ int    cmax_s[M_];
  if (threadIdx.x < 32) {
    cmax_s[threadIdx.x] = colmax[b * M_ + threadIdx.x];
    tdm_load_targets(targets + (size_t)b * M_ * 4, tgt);
  }
  __syncthreads();

  float4 a = ((const float4*)anchors)[(size_t)b * N_ + n];
  float vmax = -1.0f; int ori = 0; bool restore = false;
  for (int m = 0; m < M_; ++m) {
    float iou = iou_one(a, tgt[m]);
    if (iou > vmax) { vmax = iou; ori = m; }
    restore = restore || (__float_as_int(iou) == cmax_s[m]);  // bit-exact recompute
  }

  int ind = (vmax < BG_THR) ? -1 : ((vmax < FG_THR) ? -2 : ori);
  if (restore) ind = ori;
  float sh = sizes[b * 2 + 0], sw = sizes[b * 2 + 1];
  bool inside = (a.x >= 0.0f) && (a.y >= 0.0f) &&
                (a.z <= sw - 1.0f) && (a.w <= sh - 1.0f);
  if (!inside) ind = -2;

  if (ind >= 0) {
    int slot = atomicAdd(&pos_cnt[b], 1);
    if (slot < NUM_POS_MAX) pos_buf[b * NUM_POS_MAX + slot] = (n << 5) | ind;
  } else if (ind == -1) {
    int slot = atomicAdd(&neg_cnt[b], 1);
    if (slot < NUM_PER_IMAGE) neg_buf[b * NUM_PER_IMAGE + slot] = n;
  }
}

// ---------------- kernel 3: sampled BCE + smooth-L1, WMMA block reduction ----
__global__ void rpn_loss_kernel(const float* __restrict__ anchors,
                                const float* __restrict__ targets,
                                const float* __restrict__ logits,
                                const float* __restrict__ bregs,
                                const int* __restrict__ pos_cnt,
                                const int* __restrict__ neg_cnt,
                                const int* __restrict__ pos_buf,
                                const int* __restrict__ neg_buf,
                                float* __restrict__ accum) {
  int b = blockIdx.x;
  int t = threadIdx.x;
  int np = min(pos_cnt[b], NUM_POS_MAX);
  int nn = min(neg_cnt[b], NUM_PER_IMAGE - np);

  float cls = 0.0f, cnt = 0.0f, reg = 0.0f;
  if (t < np) {
    int packed = pos_buf[b * NUM_POS_MAX + t];
    int n = packed >> 5, m = packed & 31;
    int ai = n % A_, hw = n / A_;
    float l = logits[((size_t)b * A_ + ai) * HW_ + hw];
    cls += fmaxf(l, 0.0f) - l + log1pf(expf(-fabsf(l)));   // BCE, target 1
    cnt += 1.0f;
    float4 an = ((const float4*)anchors)[(size_t)b * N_ + n];
    float4 tg = ((const float4*)targets)[(size_t)b * M_ + m];
    float ws = an.z - an.x + 1.0f, hs = an.w - an.y + 1.0f;
    float xc = an.x + 0.5f * ws,  yc = an.y + 0.5f * hs;
    float tws = tg.z - tg.x + 1.0f, ths = tg.w - tg.y + 1.0f;
    float txc = tg.x + 0.5f * tws, tyc = tg.y + 0.5f * ths;
    float off[4] = { (txc - xc) / ws, (tyc - yc) / hs,
                     __logf(tws / ws), __logf(ths / hs) };
#pragma unroll
    for (int c = 0; c < 4; ++c) {
      float br = bregs[(((size_t)b * A_ + ai) * 4 + c) * HW_ + hw];
      float d = fabsf(br - off[c]);
      reg += (d < BETA) ? (0.5f * d * d / BETA) : (d - 0.5f * BETA);
    }
  }
  if (t < nn) {
    int n = neg_buf[b * NUM_PER_IMAGE + t];
    int ai = n % A_, hw = n / A_;
    float l = logits[((size_t)b * A_ + ai) * HW_ + hw];
    cls += fmaxf(l, 0.0f) + log1pf(expf(-fabsf(l)));       // BCE, target 0
    cnt += 1.0f;
  }

  __shared__ float red[3][256];
  red[0][t] = cls; red[1][t] = cnt; red[2][t] = reg;
  __syncthreads();

  // Tensor-core reduction: wave 0 (EXEC all-ones) sums 256 partials per
  // quantity with 4 chained V_WMMA_F32_16X16X4_F32 (ones-matrix B -> row sums
  // accumulate in C), then folds the 8 C registers + one shfl_xor(16).
  if (t < 32) {
    v2f ones; ones[0] = 1.0f; ones[1] = 1.0f;
    float tot[3];
#pragma unroll
    for (int q = 0; q < 3; ++q) {
      v8f c = {0.f, 0.f, 0.f, 0.f, 0.f, 0.f, 0.f, 0.f};
#pragma unroll
      for (int ch = 0; ch < 4; ++ch) {
        v2f av; av[0] = red[q][ch * 64 + t]; av[1] = red[q][ch * 64 + 32 + t];
        c = __builtin_amdgcn_wmma_f32_16x16x4_f32(
            /*neg_a=*/false, av, /*neg_b=*/false, ones,
            /*c_mod=*/(short)0, c, /*reuse_a=*/false, /*reuse_b=*/false);
      }
      float s = c[0] + c[1] + c[2] + c[3] + c[4] + c[5] + c[6] + c[7];
      s += __shfl_xor(s, 16, 32);
      tot[q] = s;
    }
    if (t == 0) {
      atomicAdd(&accum[0], tot[0]);
      atomicAdd(&accum[1], tot[1]);
      atomicAdd(&accum[2], tot[2]);
    }
  }
}

// ---------------- kernel 4: final divide ----------------
__global__ void rpn_final_kernel(const float* __restrict__ accum, float* __restrict__ out) {
  if (threadIdx.x == 0) {
    float total = accum[1];
    out[0] = accum[0] / total;
    out[1] = accum[2] / total;
  }
}

// ---------------- host launcher ----------------
extern "C" void kernel_launch(void* const* d_in, const int* in_sizes, int n_in,
                              void* d_out, int out_size, void* d_ws, size_t ws_size,
                              hipStream_t stream) {
  const float* anchors = (const float*)d_in[0];   // [B, N, 4]
  const float* logits  = (const float*)d_in[1];   // [B, A, H, W]
  const float* bregs   = (const float*)d_in[2];   // [B, 4A, H, W]
  const float* sizes   = (const float*)d_in[3];   // [B, 2]
  const float* targets = (const float*)d_in[4];   // [B, M, 4]
  float* out = (float*)d_out;

  // workspace carve-up (all 4-byte aligned, ~6.7 KB total)
  int* colmax  = (int*)d_ws;                      // B*M float-bit maxes
  int* pos_cnt = colmax + B_ * M_;
  int* neg_cnt = pos_cnt + B_;
  int* pos_buf = neg_cnt + B_;                    // B*128 packed (n<<5)|m
  int* neg_buf = pos_buf + B_ * NUM_POS_MAX;      // B*256 anchor indices
  float* accum = (float*)(neg_buf + B_ * NUM_PER_IMAGE);  // cls, cnt, reg

  rpn_init_kernel<<<1, 256, 0, stream>>>(colmax, pos_cnt, neg_cnt, accum);

  dim3 grid(N_ / 256, B_);   // 504 x 4 blocks, 256 threads (8 wave32s)
  rpn_colmax_kernel<<<grid, 256, 0, stream>>>(anchors, targets, colmax);
  rpn_assign_kernel<<<grid, 256, 0, stream>>>(anchors, targets, sizes, colmax,
                                              pos_cnt, neg_cnt, pos_buf, neg_buf);
  rpn_loss_kernel<<<B_, 256, 0, stream>>>(anchors, targets, logits, bregs,
                                          pos_cnt, neg_cnt, pos_buf, neg_buf, accum);
  rpn_final_kernel<<<1, 32, 0, stream>>>(accum, out);
}